// GATModule_86088324481254
// MI455X (gfx1250) — compile-verified
//
#include <hip/hip_runtime.h>
#include <hip/hip_bf16.h>

// ---------------------------------------------------------------------------
// Dense 2-layer GAT on gfx1250 (MI455X).
//   N=4096, Din=512, Dhid=512 (H=8, Dh=64), Dout=128 (H=1, Dh=128)
// fp32 end-to-end on V_WMMA_F32_16X16X4_F32 (matches reference numerics; the
// problem is adjacency-bandwidth bound so low-precision matrix ops buy
// nothing). Flash-style attention: the [N,N,H] tensor is never materialized.
// All dims are compile-time constants to strip scalar/branch overhead.
// ---------------------------------------------------------------------------

typedef __attribute__((ext_vector_type(2))) float v2f;
typedef __attribute__((ext_vector_type(8))) float v8f;

__device__ __forceinline__ v8f wmma4(v2f a, v2f b, v8f c) {
  // D = A(16x4) * B(4x16) + C(16x16), fp32.
  return __builtin_amdgcn_wmma_f32_16x16x4_f32(
      /*neg_a=*/false, a, /*neg_b=*/false, b,
      /*c_mod=*/(short)0, c, /*reuse_a=*/false, /*reuse_b=*/false);
}

__device__ __forceinline__ float lrelu(float t) { return t > 0.f ? t : 0.2f * t; }

// ---------------------------------------------------------------------------
// Z[M,NW] = X[M,K] @ W[K,NW] via 16x16x4 fp32 WMMA.
// block = 128 threads (4 waves); wave w owns column tile blockIdx.y*64 + w*16.
// A-operand: lane l holds X[m0 + (l&15)][k0 + 2*(l>>4) .. +1]  (one b64 load)
// B-operand: lane l holds W[k0 + 2*(l>>4) + {0,1}][n0 + (l&15)]
// ---------------------------------------------------------------------------
template <int K, int NW>
__global__ void __launch_bounds__(128)
gemm_wmma_f32(const float* __restrict__ X, const float* __restrict__ W,
              float* __restrict__ Z) {
  const int lane = threadIdx.x & 31;
  const int wave = threadIdx.x >> 5;
  const int t16  = lane & 15;
  const int half = lane >> 4;
  const int m0   = blockIdx.x * 16;
  const int n0   = blockIdx.y * 64 + wave * 16;

  const float* __restrict__ xrow = X + (size_t)(m0 + t16) * K;
  v8f acc = {};
#pragma unroll 4
  for (int k0 = 0; k0 < K; k0 += 4) {
    const int ka = k0 + half * 2;              // even -> 8B aligned
    v2f a = *(const v2f*)(xrow + ka);
    v2f b;
    b.x = W[(size_t)ka * NW + n0 + t16];
    b.y = W[(size_t)(ka + 1) * NW + n0 + t16];
    acc = wmma4(a, b, acc);
  }
#pragma unroll
  for (int r = 0; r < 8; ++r) {
    const int row = m0 + r + half * 8;
    Z[(size_t)row * NW + n0 + t16] = acc[r];
  }
}

// ---------------------------------------------------------------------------
// es[i,h] = z[i,h,:].a_src[h]; ed[i,h] = z[i,h,:].a_dst[h]
// ---------------------------------------------------------------------------
template <int N, int H, int Dh>
__global__ void attn_scores(const float* __restrict__ z,
                            const float* __restrict__ a_src,
                            const float* __restrict__ a_dst,
                            float* __restrict__ es, float* __restrict__ ed) {
  const int idx = blockIdx.x * blockDim.x + threadIdx.x;
  if (idx >= N * H) return;
  const int i = idx / H, h = idx % H;
  const float* __restrict__ zp = z + (size_t)i * H * Dh + (size_t)h * Dh;
  const float* __restrict__ as = a_src + (size_t)h * Dh;
  const float* __restrict__ ad = a_dst + (size_t)h * Dh;
  float s1 = 0.f, s2 = 0.f;
#pragma unroll 8
  for (int d = 0; d < Dh; ++d) {
    const float v = zp[d];
    s1 += v * as[d];
    s2 += v * ad[d];
  }
  es[idx] = s1;
  ed[idx] = s2;
}

// ---------------------------------------------------------------------------
// Pass 1: per (row i, head h) online softmax stats over neighbors j.
// One block (256 thr) per row; online (max,sum) merge in LDS.
// Masked entries use -1e9 like the reference; exp(-1e9 - m) == 0 in fp32.
// ---------------------------------------------------------------------------
template <int N, int H>
__global__ void __launch_bounds__(256)
softmax_stats(const float* __restrict__ adj, const float* __restrict__ es,
              const float* __restrict__ ed, float* __restrict__ mout,
              float* __restrict__ sout) {
  __shared__ float sm[H * 256];
  __shared__ float ss[H * 256];
  const int i = blockIdx.x;
  const int t = threadIdx.x;

  float esi[H], lm[H], ls[H];
#pragma unroll
  for (int h = 0; h < H; ++h) {
    esi[h] = es[(size_t)i * H + h];
    lm[h]  = -3.0e38f;
    ls[h]  = 0.f;
  }
  const float* __restrict__ arow = adj + (size_t)i * N;
  for (int j = t; j < N; j += 256) {
    __builtin_prefetch(arow + j + 1024, 0, 0);   // global_prefetch_b8 ahead
    const float av = arow[j];
#pragma unroll
    for (int h = 0; h < H; ++h) {
      const float e = (av > 0.f) ? lrelu(esi[h] + ed[(size_t)j * H + h]) : -1e9f;
      if (e > lm[h]) {
        ls[h] = ls[h] * __expf(lm[h] - e) + 1.f;
        lm[h] = e;
      } else {
        ls[h] += __expf(e - lm[h]);
      }
    }
  }
#pragma unroll
  for (int h = 0; h < H; ++h) {
    sm[h * 256 + t] = lm[h];
    ss[h * 256 + t] = ls[h];
  }
  __syncthreads();
  for (int st = 128; st > 0; st >>= 1) {
    if (t < st) {
#pragma unroll
      for (int h = 0; h < H; ++h) {
        const float m1 = sm[h * 256 + t], m2 = sm[h * 256 + t + st];
        const float s1 = ss[h * 256 + t], s2 = ss[h * 256 + t + st];
        const float M  = fmaxf(m1, m2);
        sm[h * 256 + t] = M;
        ss[h * 256 + t] = s1 * __expf(m1 - M) + s2 * __expf(m2 - M);
      }
    }
    __syncthreads();
  }
  if (t < H) {
    mout[(size_t)i * H + t] = sm[t * 256];
    sout[(size_t)i * H + t] = ss[t * 256];
  }
}

// ---------------------------------------------------------------------------
// Pass 2: out[i,h,:] = (1/s_i) * sum_j p_ij * z[j,h,:], p recomputed on the fly.
// grid = (N/16, H); block = 32 * Dh/(16*NT); each wave owns NT 16-col tiles so
// the P tile (adj load + 2x expf) is built ONCE per K-step and reused for NT
// WMMAs — amortizes the adjacency stream, which is the roofline limiter.
// A-operand (P tile 16x4): lane l -> rows i0+(l&15), k = 2*(l>>4)+{0,1}.
// B-operand (Z tile 4x16): lane l -> z[j0+2*(l>>4)+{0,1}][col tile + (l&15)].
// EXEC is full (all dims divide evenly; no early exits) as WMMA requires.
// ---------------------------------------------------------------------------
template <int N, int H, int Dh, int NT>
__global__ void aggregate_wmma(const float* __restrict__ adj,
                               const float* __restrict__ z,
                               const float* __restrict__ es,
                               const float* __restrict__ ed,
                               const float* __restrict__ mrow,
                               const float* __restrict__ srow,
                               float* __restrict__ out) {
  const int lane  = threadIdx.x & 31;
  const int wave  = threadIdx.x >> 5;
  const int t16   = lane & 15;
  const int half  = lane >> 4;
  const int i0    = blockIdx.x * 16;
  const int h     = blockIdx.y;
  const int ncol0 = wave * (16 * NT);
  constexpr int LD = H * Dh;

  const int   mr   = i0 + t16;                 // A-operand row for this lane
  const float es_m = es[(size_t)mr * H + h];
  const float mx_m = mrow[(size_t)mr * H + h];
  const float* __restrict__ adjrow = adj + (size_t)mr * N;

  int col[NT];
#pragma unroll
  for (int t = 0; t < NT; ++t) col[t] = h * Dh + ncol0 + t * 16 + t16;

  v8f acc[NT];
#pragma unroll
  for (int t = 0; t < NT; ++t) acc[t] = v8f{};

#pragma unroll 4
  for (int j0 = 0; j0 < N; j0 += 4) {
    const int ja = j0 + half * 2;              // even -> 8B aligned
    __builtin_prefetch(adjrow + ja + 256, 0, 0);  // global_prefetch_b8 ahead
    const v2f a2 = *(const v2f*)(adjrow + ja); // adj[mr][ja], adj[mr][ja+1]
    const float eda = ed[(size_t)ja * H + h];
    const float edb = ed[(size_t)(ja + 1) * H + h];
    v2f A;
    A.x = (a2.x > 0.f) ? __expf(lrelu(es_m + eda) - mx_m) : 0.f;
    A.y = (a2.y > 0.f) ? __expf(lrelu(es_m + edb) - mx_m) : 0.f;
#pragma unroll
    for (int t = 0; t < NT; ++t) {
      v2f B;
      B.x = z[(size_t)ja * LD + col[t]];
      B.y = z[(size_t)(ja + 1) * LD + col[t]];
      acc[t] = wmma4(A, B, acc[t]);
    }
  }
#pragma unroll
  for (int r = 0; r < 8; ++r) {
    const int row = i0 + r + half * 8;
    const float inv = 1.f / srow[(size_t)row * H + h];
#pragma unroll
    for (int t = 0; t < NT; ++t)
      out[(size_t)row * LD + h * Dh + ncol0 + t * 16 + t16] = acc[t][r] * inv;
  }
}

// ---------------------------------------------------------------------------
// ELU in place.
// ---------------------------------------------------------------------------
__global__ void elu_kernel(float* __restrict__ p, int n) {
  const int i = blockIdx.x * blockDim.x + threadIdx.x;
  if (i >= n) return;
  const float v = p[i];
  p[i] = v > 0.f ? v : (__expf(v) - 1.f);
}

// ---------------------------------------------------------------------------
extern "C" void kernel_launch(void* const* d_in, const int* in_sizes, int n_in,
                              void* d_out, int out_size, void* d_ws, size_t ws_size,
                              hipStream_t stream) {
  (void)in_sizes; (void)n_in; (void)out_size; (void)ws_size;
  constexpr int N = 4096, Din = 512, Dhid = 512, Dout = 128;
  constexpr int H1 = 8, Dh1 = 64, H2 = 1, Dh2 = 128;

  const float* x   = (const float*)d_in[0];
  const float* adj = (const float*)d_in[1];
  const float* W1  = (const float*)d_in[2];
  const float* a1s = (const float*)d_in[3];
  const float* a1d = (const float*)d_in[4];
  const float* W2  = (const float*)d_in[5];
  const float* a2s = (const float*)d_in[6];
  const float* a2d = (const float*)d_in[7];
  float* outp = (float*)d_out;

  float* ws  = (float*)d_ws;                      // ~18.6 MB used
  float* z1  = ws;                                // N*Dhid
  float* hb  = z1  + (size_t)N * Dhid;            // N*Dhid
  float* z2  = hb  + (size_t)N * Dhid;            // N*Dout
  float* es1 = z2  + (size_t)N * Dout;            // N*H1
  float* ed1 = es1 + (size_t)N * H1;
  float* m1  = ed1 + (size_t)N * H1;
  float* s1  = m1  + (size_t)N * H1;
  float* es2 = s1  + (size_t)N * H1;              // N
  float* ed2 = es2 + (size_t)N;
  float* m2  = ed2 + (size_t)N;
  float* s2  = m2  + (size_t)N;

  // ---- Layer 1 ----
  gemm_wmma_f32<Din, Dhid><<<dim3(N / 16, Dhid / 64), 128, 0, stream>>>(x, W1, z1);
  attn_scores<N, H1, Dh1><<<(N * H1) / 256, 256, 0, stream>>>(z1, a1s, a1d, es1, ed1);
  softmax_stats<N, H1><<<N, 256, 0, stream>>>(adj, es1, ed1, m1, s1);
  aggregate_wmma<N, H1, Dh1, 2><<<dim3(N / 16, H1), 32 * (Dh1 / 32), 0, stream>>>(
      adj, z1, es1, ed1, m1, s1, hb);
  elu_kernel<<<(N * Dhid) / 256, 256, 0, stream>>>(hb, N * Dhid);

  // ---- Layer 2 ----
  gemm_wmma_f32<Dhid, Dout><<<dim3(N / 16, Dout / 64), 128, 0, stream>>>(hb, W2, z2);
  attn_scores<N, H2, Dh2><<<(N * H2 + 255) / 256, 256, 0, stream>>>(z2, a2s, a2d, es2, ed2);
  softmax_stats<N, H2><<<N, 256, 0, stream>>>(adj, es2, ed2, m2, s2);
  aggregate_wmma<N, H2, Dh2, 2><<<dim3(N / 16, H2), 32 * (Dh2 / 32), 0, stream>>>(
      adj, z2, es2, ed2, m2, s2, outp);
}